// MoEDepthwiseConv2d_38543036514532
// MI455X (gfx1250) — compile-verified
//
#include <hip/hip_runtime.h>
#include <stdint.h>

// Problem dims (fixed by the reference)
#define Bn 32
#define Cn 256
#define Hn 64
#define Wn 64
#define En 4
#define Kn 5
#define CK (Cn * Kn * Kn)   // 6400 = per-expert flattened filter bank width

typedef __attribute__((ext_vector_type(2))) float    v2f;
typedef __attribute__((ext_vector_type(4))) float    v4f;
typedef __attribute__((ext_vector_type(8))) float    v8f;
typedef __attribute__((ext_vector_type(4))) uint32_t v4u;
typedef __attribute__((ext_vector_type(8))) uint32_t v8u;

// ---------------------------------------------------------------------------
// Stage 1: agg[32,6400] = routing[32,4] x weight[4,6400] using
// V_WMMA_F32_16X16X4_F32 (native fp32 WMMA, K=4 matches E=4 exactly).
// One wave per 16x16 output tile. grid = 2 (M tiles) * 400 (N tiles).
// ---------------------------------------------------------------------------
__global__ __launch_bounds__(32) void moe_agg_wmma(
    const float* __restrict__ routing,   // [32,4]
    const float* __restrict__ weight,    // [4,6400] (E,C*K*K)
    float* __restrict__ agg)             // [32,6400]
{
    const int wid  = blockIdx.x;
    const int mt   = wid & 1;        // M tile: 0..1
    const int nt   = wid >> 1;       // N tile: 0..399
    const int lane = threadIdx.x;    // wave32
    const int half = lane >> 4;      // 0: lanes 0-15, 1: lanes 16-31
    const int l15  = lane & 15;

    // A (16x4 f32, 2 VGPRs): lane%16 = M row; lanes<16 hold K=0,1; lanes>=16 hold K=2,3
    const int k0 = half * 2;
    const int m  = mt * 16 + l15;
    v2f a;
    a.x = routing[m * En + k0];
    a.y = routing[m * En + k0 + 1];

    // B (4x16 f32, 2 VGPRs): lane%16 = N col; lanes<16 hold K=0,1; lanes>=16 hold K=2,3
    const int n = nt * 16 + l15;
    v2f b;
    b.x = weight[(k0    ) * CK + n];
    b.y = weight[(k0 + 1) * CK + n];

    v8f c = {};
    // 8 args: (neg_a, A, neg_b, B, c_mod, C, reuse_a, reuse_b)
    v8f d = __builtin_amdgcn_wmma_f32_16x16x4_f32(
        false, a, false, b, (short)0, c, false, false);

    // D layout (16x16 f32): VGPR v holds M = v (lanes 0-15) or M = v+8 (lanes 16-31)
#pragma unroll
    for (int v = 0; v < 8; ++v) {
        const int mo = mt * 16 + v + half * 8;
        agg[mo * CK + n] = d[v];
    }
}

// ---------------------------------------------------------------------------
// Stage 2: depthwise 5x5 conv, one (b,c) 64x64 plane per 256-thread block.
// LDS tile 68 rows x 72 floats; halo of 2 zero-filled on all sides.
// The 64x64 plane is DMA'd into the tile interior by the Tensor Data Mover:
// one TENSOR_LOAD_TO_LDS per block, with the TDM pad feature (64 DWORDs data,
// 8 DWORDs pad) producing the 72-float LDS row stride in hardware.
// Each thread then produces a 1x16 output strip; per filter row it pulls a
// 20-float window from LDS once and runs 5x16 register FMAs.
// Traffic: x read once (TDM), out written once (NT stores) -> bandwidth-optimal.
// ---------------------------------------------------------------------------
#define LDS_H 68
#define LDS_W 72

__global__ __launch_bounds__(256) void moe_dwconv(
    const float* __restrict__ x,     // [B,C,64,64]
    const float* __restrict__ agg,   // [B, C*25] aggregated filters
    float* __restrict__ out)         // [B,C,64,64]
{
    __shared__ float tile[LDS_H * LDS_W];

    const int bc  = blockIdx.x;      // b*C + c
    const int tid = threadIdx.x;

    // Zero the whole tile (covers the halo).
    for (int i = tid; i < LDS_H * LDS_W; i += 256) tile[i] = 0.0f;

    // Per-plane filter taps (uniform across the block).
    float w[25];
    {
        const float* wp = agg + bc * 25;
#pragma unroll
        for (int j = 0; j < 25; ++j) w[j] = wp[j];
    }

    __syncthreads();  // zero-fill complete before the TDM writes the interior

    // --- TDM: DMA the 64x64 fp32 plane into the tile interior -------------
    // Per-wave instruction (EXEC ignored); issue from wave 0 only -> one DMA.
    if (tid < 32) {
        const uint64_t ga = (uint64_t)(uintptr_t)(x + (size_t)bc * (Hn * Wn));
        // Generic pointer to LDS: low 32 bits are the byte offset within the
        // workgroup's LDS allocation (what D#.lds_addr wants).
        const uint32_t lds_base =
            (uint32_t)(uintptr_t)(&tile[2 * LDS_W + 4]);

        // D# Group 0 (128b): count=1 | lds_addr | global_addr | type=2
        v4u g0;
        g0[0] = 0x1u;                                // count=1, user mode
        g0[1] = lds_base;                            // lds_addr (bytes)
        g0[2] = (uint32_t)ga;                        // global_addr[31:0]
        g0[3] = (uint32_t)(ga >> 32) | (2u << 30);   // global_addr[56:32] | type=2

        // D# Group 1 (256b):
        //  d0: wg_mask=0 | data_size=2 (4B) | pad_enable | pad_interval=5 (64DW)
        //      | pad_amount=7 (8DW)
        //  d1: atomic_barrier_addr=0 | tensor_dim0.lo16 (64)
        //  d2: tensor_dim0.hi16 (0)  | tensor_dim1.lo16 (64)
        //  d3: tensor_dim1.hi16 (0)  | tile_dim0 (64)
        //  d4: tile_dim1 (64)        | tile_dim2 (0)
        //  d5: tensor_dim0_stride[31:0] (64)
        //  d6: tensor_dim0_stride[47:32] | tensor_dim1_stride.lo16 (0)
        //  d7: tensor_dim1_stride[47:16] (0)
        v8u g1;
        g1[0] = (2u << 16) | (1u << 20) | (5u << 22) | (7u << 25);
        g1[1] = (uint32_t)Wn << 16;
        g1[2] = (uint32_t)Hn << 16;
        g1[3] = (uint32_t)Wn << 16;
        g1[4] = (uint32_t)Hn;
        g1[5] = (uint32_t)Wn;
        g1[6] = 0u;
        g1[7] = 0u;

        // D# Groups 2/3: zeros (2D tensor, no iteration, no gather)
        v4u g2 = {0u, 0u, 0u, 0u};
        v4u g3 = {0u, 0u, 0u, 0u};

        asm volatile("tensor_load_to_lds %0, %1, %2, %3"
                     :
                     : "s"(g0), "s"(g1), "s"(g2), "s"(g3)
                     : "memory");
        __builtin_amdgcn_s_wait_tensorcnt(0);
    }

    __syncthreads();  // TDM data visible to all waves

    // Compute: thread -> output row r, columns x0..x0+15.
    const int r  = tid >> 2;         // 0..63
    const int x0 = (tid & 3) * 16;   // 0,16,32,48

    float acc[16];
#pragma unroll
    for (int i = 0; i < 16; ++i) acc[i] = 0.0f;

#pragma unroll
    for (int ky = 0; ky < Kn; ++ky) {
        // 20-float sliding window for this filter row.
        float row[20];
        const float* rp = &tile[(r + ky) * LDS_W + x0 + 2];
#pragma unroll
        for (int j = 0; j < 20; ++j) row[j] = rp[j];

#pragma unroll
        for (int kx = 0; kx < Kn; ++kx) {
            const float wv = w[ky * Kn + kx];
#pragma unroll
            for (int xx = 0; xx < 16; ++xx)
                acc[xx] = fmaf(wv, row[xx + kx], acc[xx]);
        }
    }

    // Output is written once and never re-read: stream it (non-temporal b128).
    float* op = out + (size_t)bc * (Hn * Wn) + r * Wn + x0;
#pragma unroll
    for (int q = 0; q < 4; ++q) {
        v4f v = { acc[4 * q], acc[4 * q + 1], acc[4 * q + 2], acc[4 * q + 3] };
        __builtin_nontemporal_store(v, (v4f*)(op + 4 * q));
    }
}

// ---------------------------------------------------------------------------
extern "C" void kernel_launch(void* const* d_in, const int* in_sizes, int n_in,
                              void* d_out, int out_size, void* d_ws, size_t ws_size,
                              hipStream_t stream) {
    const float* x       = (const float*)d_in[0];  // [32,256,64,64]
    const float* routing = (const float*)d_in[1];  // [32,4]
    const float* weight  = (const float*)d_in[2];  // [4,256,1,5,5]
    float*       out     = (float*)d_out;          // [32,256,64,64]

    // Scratch: aggregated per-sample filters, 32*6400 floats = 800 KB.
    float* agg = (float*)d_ws;

    // Stage 1: WMMA aggregation (800 one-wave tiles).
    moe_agg_wmma<<<dim3(2 * (CK / 16)), dim3(32), 0, stream>>>(routing, weight, agg);

    // Stage 2: TDM-staged streaming depthwise conv, one block per (b,c) plane.
    moe_dwconv<<<dim3(Bn * Cn), dim3(256), 0, stream>>>(x, agg, out);
}